// Spatial_Weighting_57964878627432
// MI455X (gfx1250) — compile-verified
//
#include <hip/hip_runtime.h>
#include <hip/hip_bf16.h>
#include <math.h>

// Problem dims (from reference): qkv1 (3, B*T=32, H=12, N=576, D=64) f32
#define BT   32
#define Hh   12
#define Nn   576
#define Dd   64
#define Cc   (Hh*Dd)          // 768
#define TILES (Nn/16)         // 36
#define NITER 300

typedef float v2f __attribute__((ext_vector_type(2)));
typedef float v8f __attribute__((ext_vector_type(8)));

static __device__ __forceinline__ long hnd() { return (long)Hh*Nn*Dd; }

// ---------------------------------------------------------------------------
// K1: feats[bt][n][m] = sum_c key1[bt][n][c] * query1[bt][m][c]
// Exact f32 GEMM via V_WMMA_F32_16X16X4_F32. One wave -> one 16x16 tile.
// A frag (16x4): lane<16 -> m=lane, k={0,1}; lane>=16 -> m=lane-16, k={2,3}
// B frag (4x16): lane<16 -> n=lane, k={0,1}; lane>=16 -> n=lane-16, k={2,3}
// D (16x16 f32): vgpr r, lane<16 -> (m=r, n=lane); lane>=16 -> (m=r+8, n=lane-16)
// ---------------------------------------------------------------------------
__global__ __launch_bounds__(256)
void feats_wmma(const float* __restrict__ qkv1, float* __restrict__ F) {
  const int bt   = blockIdx.y;
  const int wave = threadIdx.x >> 5;
  const int lane = threadIdx.x & 31;
  const int tile = blockIdx.x * 8 + wave;        // 0..1295
  const int ti = tile / TILES, tj = tile % TILES;

  const float* Kb = qkv1 + (long)BT * hnd() + (long)bt * hnd();  // key1   (qkv1[1])
  const float* Qb = qkv1 + (long)bt * hnd();                     // query1 (qkv1[0])

  const int ml = lane & 15;
  const int kb = (lane >> 4) << 1;               // 0 or 2
  const long rowA = (long)(ti * 16 + ml) * Dd;   // key row (feats row index n)
  const long rowB = (long)(tj * 16 + ml) * Dd;   // query row (feats col index m)

  v8f acc = {};
  for (int kc = 0; kc < Cc; kc += 4) {
    const long hoff = (long)(kc >> 6) * Nn * Dd;
    const int  dlo  = (kc & 63) + kb;            // even -> 8B aligned
    v2f a = *(const v2f*)(Kb + hoff + rowA + dlo);
    v2f b = *(const v2f*)(Qb + hoff + rowB + dlo);
    acc = __builtin_amdgcn_wmma_f32_16x16x4_f32(
        /*neg_a=*/false, a, /*neg_b=*/false, b,
        /*c_mod=*/(short)0, acc, /*reuse_a=*/false, /*reuse_b=*/false);
  }

  float* Fb = F + (long)bt * Nn * Nn;
  const int ncol  = tj * 16 + (lane & 15);
  const int mbase = ti * 16 + ((lane >> 4) << 3);
#pragma unroll
  for (int r = 0; r < 8; ++r)
    Fb[(long)(mbase + r) * Nn + ncol] = acc[r];
}

// ---------------------------------------------------------------------------
// K2: d_i[n] = sum_m (feats[n][m] > 0.2 ? 1 : 1e-5);  store 1/sqrt(d_i)
// ---------------------------------------------------------------------------
__global__ __launch_bounds__(Nn)
void rowsum_dinv(const float* __restrict__ F, float* __restrict__ dinv) {
  const int bt = blockIdx.x, n = threadIdx.x;
  const float* row = F + (long)bt * Nn * Nn + (long)n * Nn;
  float s = 0.0f;
  for (int m = 0; m < Nn; m += 4) {
    float4 v = *(const float4*)(row + m);
    s += (v.x > 0.2f ? 1.0f : 1e-5f);
    s += (v.y > 0.2f ? 1.0f : 1e-5f);
    s += (v.z > 0.2f ? 1.0f : 1e-5f);
    s += (v.w > 0.2f ? 1.0f : 1e-5f);
  }
  dinv[bt * Nn + n] = rsqrtf(s);
}

// ---------------------------------------------------------------------------
// K3: in-place, symmetric: B = I + S~,  S~[n][m] = dinv[n]*0.5*(W[n][m]+W[m][n])*dinv[m]
// Pair {n,m} owned exclusively by block row n = min(n,m): race-free.
// ---------------------------------------------------------------------------
__global__ __launch_bounds__(Nn)
void build_B(float* __restrict__ F, const float* __restrict__ dinv) {
  const int bt = blockIdx.x, n = blockIdx.y, m = threadIdx.x;
  if (m < n) return;
  float* Fb = F + (long)bt * Nn * Nn;
  const float fnm = Fb[(long)n * Nn + m];
  const float fmn = Fb[(long)m * Nn + n];
  const float wnm = fnm > 0.2f ? 1.0f : 1e-5f;
  const float wmn = fmn > 0.2f ? 1.0f : 1e-5f;
  float v = 0.5f * (wnm + wmn) * dinv[bt * Nn + n] * dinv[bt * Nn + m];
  if (m == n) v += 1.0f;
  Fb[(long)n * Nn + m] = v;
  Fb[(long)m * Nn + n] = v;
}

// ---------------------------------------------------------------------------
// K4: 2-vector subspace iteration on B (L2-resident: 32 matrices = 42.5MB < 192MB L2),
// then eigenvec = dinv * v1, bipartition + masked softmax -> out (bt, n).
// One block per bt, 576 threads (18 wave32).
// ---------------------------------------------------------------------------
__global__ __launch_bounds__(Nn)
void fiedler(const float* __restrict__ Bm, const float* __restrict__ dinv,
             float* __restrict__ out) {
  __shared__ float sv0[Nn], sv1[Nn];
  __shared__ float r0[1024], r1[1024], r2[1024];
  const int bt = blockIdx.x, n = threadIdx.x;
  const float* Brow = Bm + (long)bt * Nn * Nn + (long)n * Nn;

  // deterministic init: v0 ~ smooth, v1 ~ pseudo-random (overlap with Fiedler dir)
  sv0[n] = 1.0f;
  unsigned h = (unsigned)n * 2654435761u + 12345u;
  sv1[n] = (float)((h >> 9) & 0xFFFF) * (1.0f / 32768.0f) - 1.0f;
  if (n < 1024 - Nn) { r0[Nn + n] = 0.0f; r1[Nn + n] = 0.0f; r2[Nn + n] = 0.0f; }
  __syncthreads();

  float y0 = 0.0f, y1 = 0.0f;
  for (int it = 0; it < NITER; ++it) {
    y0 = 0.0f; y1 = 0.0f;
    for (int m = 0; m < Nn; m += 4) {
      float4 bv = *(const float4*)(Brow + m);
      y0 += bv.x * sv0[m] + bv.y * sv0[m + 1] + bv.z * sv0[m + 2] + bv.w * sv0[m + 3];
      y1 += bv.x * sv1[m] + bv.y * sv1[m + 1] + bv.z * sv1[m + 2] + bv.w * sv1[m + 3];
    }
    __syncthreads();
    r0[n] = y0 * y0; r1[n] = y0 * y1; r2[n] = y1 * y1;
    __syncthreads();
    for (int s = 512; s > 0; s >>= 1) {
      if (n < s) { r0[n] += r0[n + s]; r1[n] += r1[n + s]; r2[n] += r2[n + s]; }
      __syncthreads();
    }
    const float n00 = fmaxf(r0[0], 1e-30f);
    const float d01 = r1[0];
    const float n11 = r2[0];
    const float inv0 = rsqrtf(n00);
    const float c    = d01 / n00;                 // Gram-Schmidt coefficient
    const float y1o  = y1 - c * y0;
    const float inv1 = rsqrtf(fmaxf(n11 - d01 * c, 1e-30f));
    __syncthreads();
    sv0[n] = y0 * inv0;
    sv1[n] = y1o * inv1;
    __syncthreads();
  }

  // eigenvec = dinv * u  (u = 2nd eigenvector; sign ambiguity cancels below)
  const float e = dinv[bt * Nn + n] * sv1[n];

  // avg = mean(e)
  r0[n] = e;
  if (n < 1024 - Nn) r0[Nn + n] = 0.0f;
  __syncthreads();
  for (int s = 512; s > 0; s >>= 1) { if (n < s) r0[n] += r0[n + s]; __syncthreads(); }
  const float avg = r0[0] * (1.0f / Nn);
  __syncthreads();

  // included iff (avg>0) == (e>0); ev = |e| on included, -inf elsewhere
  const float val = ((avg > 0.0f) == (e > 0.0f)) ? fabsf(e) : -INFINITY;

  // max reduce
  r0[n] = val;
  if (n < 1024 - Nn) r0[Nn + n] = -INFINITY;
  __syncthreads();
  for (int s = 512; s > 0; s >>= 1) { if (n < s) r0[n] = fmaxf(r0[n], r0[n + s]); __syncthreads(); }
  const float mx = r0[0];
  __syncthreads();

  // sum of exp
  const float ex = expf(val - mx);   // expf(-inf) == 0
  r0[n] = ex;
  if (n < 1024 - Nn) r0[Nn + n] = 0.0f;
  __syncthreads();
  for (int s = 512; s > 0; s >>= 1) { if (n < s) r0[n] += r0[n + s]; __syncthreads(); }
  out[bt * Nn + n] = ex / r0[0];
}

// ---------------------------------------------------------------------------
extern "C" void kernel_launch(void* const* d_in, const int* in_sizes, int n_in,
                              void* d_out, int out_size, void* d_ws, size_t ws_size,
                              hipStream_t stream) {
  (void)in_sizes; (void)n_in; (void)out_size; (void)ws_size;
  const float* qkv1 = (const float*)d_in[0];   // qkv2 (d_in[1]) unused by reference
  float* F    = (float*)d_ws;                  // 32*576*576 f32 = 42.5 MB
  float* dinv = F + (size_t)BT * Nn * Nn;      // 32*576 f32
  float* out  = (float*)d_out;                 // (4,8,576) f32

  feats_wmma <<<dim3(TILES * TILES / 8, BT), 256, 0, stream>>>(qkv1, F);
  rowsum_dinv<<<BT, Nn, 0, stream>>>(F, dinv);
  build_B    <<<dim3(BT, Nn), Nn, 0, stream>>>(F, dinv);
  fiedler    <<<BT, Nn, 0, stream>>>(F, dinv, out);
}